// DendriticMatcherBank_65352222376341
// MI455X (gfx1250) — compile-verified
//
#include <hip/hip_runtime.h>

// DendriticMatcherBank fused kernel for gfx1250 (MI455X).
// B=16384, K=64, T=128. ~1.08 GB of stores (46us HBM floor @23.3TB/s);
// leaky-integrator scan done as triangular GEMM on V_WMMA_F32_16X16X4_F32.
// Round 2: hoist the masked-decay B-fragments into a per-workgroup LDS table
// (they depend only on delta = t0-4i, 32 distinct values) so the WMMA inner
// loop is just ds_load_b64(A) + ds_load_b64(B) + v_wmma. Use raw v_exp_f32.

typedef __attribute__((ext_vector_type(2))) float v2f;
typedef __attribute__((ext_vector_type(4))) float f4;
typedef __attribute__((ext_vector_type(8))) float v8f;

#define B_N 16384
#define K_N 64
#define T_N 128

#define ALPHA          0.05f                   // DT / TAU_MEM
#define LOG2_1MA      -0.07400058144377693f    // log2(1 - ALPHA)
#define PSP_SCALE      0.03606737602222409f    // log2(e)/TAU_PSP (=40)
#define LOG2E          1.4426950408889634f     // softmax: TAU_SOFT = 1

static __device__ __forceinline__ float fast_exp2(float x) {
    return __builtin_amdgcn_exp2f(x);          // v_exp_f32 (args always in range)
}

__global__ __launch_bounds__(128) void dendritic_matcher_kernel(
    const float* __restrict__ spike_times,  // (B, T)
    const float* __restrict__ w,            // (K, T)
    const float* __restrict__ delays,       // (K, T)
    float* __restrict__ out)                // arrivals | psp | matcher
{
    __shared__ __align__(16) float I_lds[16][132]; // padded: banks (4m+s)%64 unique
    __shared__ __align__(16) float Btab[32][64];   // B frags: [delta_idx][2*lane+{0,1}]
    __shared__ float decay[T_N];                   // alpha*(1-alpha)^d
    __shared__ float part_sum[4][16];
    __shared__ float part_tsum[4][16];

    const int tid  = threadIdx.x;
    const int lane = tid & 31;
    const int wave = tid >> 5;

    const int r0 = blockIdx.x * 16;     // first flat row (b*K + k)
    const int b  = r0 / K_N;            // 16 | K -> whole tile shares b
    const int k0 = r0 % K_N;

    // decay LUT: one entry per thread (blockDim == 128 == T)
    decay[tid] = ALPHA * fast_exp2((float)tid * LOG2_1MA);
    __syncthreads();

    // ---------------- Phase 0: precompute masked-decay B fragments ----------
    // For k-step i of tile j: delta = 16j - 4i in {-12,-8,...,112} -> 32 values,
    // delta_idx = (delta+12)/4. Lane ln: n = ln&15, half = ln>>4,
    //   bb.x (kk=2*half)   -> d0 = delta + n - 2*half
    //   bb.y (kk=2*half+1) -> d1 = d0 - 1            (d <= 127 always)
    #pragma unroll
    for (int e = tid; e < 32 * 32; e += 128) {
        const int dix = e >> 5;
        const int ln  = e & 31;
        const int d0  = (4 * dix - 12) + (ln & 15) - 2 * (ln >> 4);
        const int d1  = d0 - 1;
        const float v0 = decay[d0 > 0 ? d0 : 0];
        const float v1 = decay[d1 > 0 ? d1 : 0];
        Btab[dix][2 * ln]     = (d0 >= 0) ? v0 : 0.0f;
        Btab[dix][2 * ln + 1] = (d1 >= 0) ? v1 : 0.0f;
    }

    float* arr_out   = out;
    float* psp_out   = out + (size_t)B_N * K_N * T_N;
    float* match_out = out + 2ull * (size_t)B_N * K_N * T_N;

    // ---------------- Phase 1: elementwise + rowmax + I into LDS ------------
    const int t4 = lane * 4;            // this lane covers t4..t4+3
    const f4 s4 = *(const f4*)(spike_times + (size_t)b * T_N + t4);

    #pragma unroll
    for (int q = 0; q < 4; ++q) {
        const int row = wave * 4 + q;   // local row 0..15
        const int k   = k0 + row;
        const f4 d4 = *(const f4*)(delays + (size_t)k * T_N + t4);
        const f4 w4 = *(const f4*)(w      + (size_t)k * T_N + t4);

        f4 arr;
        arr.x = s4.x + d4.x;  arr.y = s4.y + d4.y;
        arr.z = s4.z + d4.z;  arr.w = s4.w + d4.w;

        const size_t base = (size_t)(b * K_N + k) * T_N + t4;
        *(f4*)(arr_out + base) = arr;                       // global_store_b128

        // row max over all 128 t (32 lanes x 4)
        float m = fmaxf(fmaxf(arr.x, arr.y), fmaxf(arr.z, arr.w));
        #pragma unroll
        for (int off = 16; off >= 1; off >>= 1)
            m = fmaxf(m, __shfl_xor(m, off, 32));

        f4 psp;
        psp.x = fast_exp2((arr.x - m) * PSP_SCALE);
        psp.y = fast_exp2((arr.y - m) * PSP_SCALE);
        psp.z = fast_exp2((arr.z - m) * PSP_SCALE);
        psp.w = fast_exp2((arr.w - m) * PSP_SCALE);
        *(f4*)(psp_out + base) = psp;                       // global_store_b128

        f4 I;
        I.x = w4.x * psp.x;  I.y = w4.y * psp.y;
        I.z = w4.z * psp.z;  I.w = w4.w * psp.w;
        *(f4*)(&I_lds[row][t4]) = I;                        // ds_store_b128
    }
    __syncthreads();

    // ---------------- Phase 2: scan as triangular GEMM via WMMA -------------
    // V[m, t] = sum_{s<=t} decay[t-s] * I[m, s]
    // A 16x4 (I rows x s-block) from I_lds; B 4x16 straight from Btab.
    // Iterate k-steps in reverse so Btab index == loop counter.
    const int m_row = lane & 15;
    const int half  = lane >> 4;

    float esum[8], etsum[8];
    #pragma unroll
    for (int p = 0; p < 8; ++p) { esum[p] = 0.0f; etsum[p] = 0.0f; }

    #pragma unroll
    for (int which = 0; which < 2; ++which) {
        const int j    = (which == 0) ? wave : (7 - wave);  // load-balanced pairing
        const int tcol = j * 16 + (lane & 15);              // this lane's t (N dim)

        v8f acc = {};
        const int ksteps = 4 * j + 4;                       // skip zero upper triangle
        const int s_base = 16 * j + 12 + 2 * half;          // s_a = s_base - 4*idx
        for (int idx = 0; idx < ksteps; ++idx) {
            const v2f a  = *(const v2f*)&I_lds[m_row][s_base - 4 * idx]; // ds_load_b64
            const v2f bb = *(const v2f*)&Btab[idx][2 * lane];            // ds_load_b64
            acc = __builtin_amdgcn_wmma_f32_16x16x4_f32(
                false, a, false, bb, (short)0, acc, false, false);
        }

        // ---------------- Phase 3a: softmax partials from accumulator -------
        // C/D layout: VGPR p -> row (p + 8*half), lanes 0-15 / 16-31 -> t = tcol
        const float tf = (float)tcol;
        #pragma unroll
        for (int p = 0; p < 8; ++p) {
            const float e = fast_exp2(acc[p] * LOG2E);
            esum[p]  += e;
            etsum[p] += e * tf;
        }
    }

    // reduce over the 16 lanes of each half (xor masks stay within the half)
    #pragma unroll
    for (int p = 0; p < 8; ++p) {
        #pragma unroll
        for (int off = 8; off >= 1; off >>= 1) {
            esum[p]  += __shfl_xor(esum[p],  off, 32);
            etsum[p] += __shfl_xor(etsum[p], off, 32);
        }
    }
    if ((lane & 15) == 0) {
        #pragma unroll
        for (int p = 0; p < 8; ++p) {
            const int row = p + 8 * half;
            part_sum[wave][row]  = esum[p];
            part_tsum[wave][row] = etsum[p];
        }
    }
    __syncthreads();

    // ---------------- Phase 3b: deterministic cross-wave combine ------------
    if (tid < 16) {
        const float s  = ((part_sum[0][tid]  + part_sum[1][tid]) +
                          (part_sum[2][tid]  + part_sum[3][tid]));
        const float ts = ((part_tsum[0][tid] + part_tsum[1][tid]) +
                          (part_tsum[2][tid] + part_tsum[3][tid]));
        match_out[(size_t)b * K_N + (k0 + tid)] = ts / s;
    }
}

extern "C" void kernel_launch(void* const* d_in, const int* in_sizes, int n_in,
                              void* d_out, int out_size, void* d_ws, size_t ws_size,
                              hipStream_t stream) {
    (void)in_sizes; (void)n_in; (void)out_size; (void)d_ws; (void)ws_size;
    const float* spike_times = (const float*)d_in[0];
    const float* w           = (const float*)d_in[1];
    const float* delays      = (const float*)d_in[2];
    float* out = (float*)d_out;

    const int n_blocks = (B_N * K_N) / 16;   // 65536 workgroups, 16 rows each
    dendritic_matcher_kernel<<<n_blocks, 128, 0, stream>>>(spike_times, w, delays, out);
}